// LocalL1Loss_89421219103074
// MI455X (gfx1250) — compile-verified
//
#include <hip/hip_runtime.h>
#include <hip/hip_bf16.h>

// ---------------------------------------------------------------------------
// LocalL1Loss on MI455X (gfx1250, wave32).
// inputs/targets: (16, 3, 512, 512) fp32. Output: scalar fp32 mean.
// Kernel 1: LDS-tiled 7x7 shifted-L1 min + per-block sum (WMMA f32 reduction).
// Kernel 2: deterministic final reduction of block partials (WMMA f32).
// ---------------------------------------------------------------------------

typedef __attribute__((ext_vector_type(2))) float v2f;
typedef __attribute__((ext_vector_type(8))) float v8f;

#define NB   16
#define NC   3
#define HH   512
#define WW   512
#define HWSZ (HH * WW)

#define TX   32
#define TY   8
#define HALO 3
#define LW   (TX + 2 * HALO)   // 38
#define LH   (TY + 2 * HALO)   // 14

// Sum of one wave32's `v` across all 32 lanes, returned on every lane.
// Uses V_WMMA_F32_16X16X4_F32 with B = all-ones:
//   D[m][n] = sum_k A[m][k]  (row-sums of A, replicated over columns).
// Each lane contributes exactly one A element (a.y = 0), so
//   sum over C rows 0..7  (lanes 0..15)  = rowsums of M=0..7
//   sum over C rows 8..15 (lanes 16..31) = rowsums of M=8..15
// and the two halves combine with a single shfl_xor(16).
// EXEC must be all-ones (call uniformly from a full wave).
__device__ __forceinline__ float wave32_sum_wmma(float v) {
  v2f a; a.x = v;    a.y = 0.0f;
  v2f b; b.x = 1.0f; b.y = 1.0f;   // all-ones 4x16 B
  v8f c = {};
  c = __builtin_amdgcn_wmma_f32_16x16x4_f32(
      /*neg_a=*/false, a, /*neg_b=*/false, b,
      /*c_mod=*/(short)0, c, /*reuse_a=*/false, /*reuse_b=*/false);
  float s = c[0] + c[1] + c[2] + c[3] + c[4] + c[5] + c[6] + c[7];
  s += __shfl_xor(s, 16, 32);      // combine M=0..7 half with M=8..15 half
  return s;
}

__global__ void __launch_bounds__(TX * TY)
local_l1_min_kernel(const float* __restrict__ inputs,
                    const float* __restrict__ targets,
                    float* __restrict__ block_partials) {
  __shared__ float tile[NC][LH][LW];       // 3*14*38 floats = 6384 B
  __shared__ float wpart[TX * TY / 32];

  const int tid = threadIdx.y * TX + threadIdx.x;   // 0..255
  const int n   = blockIdx.z;
  const int gx0 = blockIdx.x * TX - HALO;
  const int gy0 = blockIdx.y * TY - HALO;

  // Stage target tile (with zero-padded halo) into LDS: 1596 elements / 256 thr.
  for (int idx = tid; idx < NC * LH * LW; idx += TX * TY) {
    int c   = idx / (LH * LW);
    int rem = idx - c * (LH * LW);
    int ly  = rem / LW;
    int lx  = rem - ly * LW;
    int gy  = gy0 + ly;
    int gx  = gx0 + lx;
    float v = 0.0f;
    if (gy >= 0 && gy < HH && gx >= 0 && gx < WW)
      v = targets[(n * NC + c) * HWSZ + gy * WW + gx];
    tile[c][ly][lx] = v;
  }
  __syncthreads();

  // Per-pixel: min over 49 shifts of channel-summed |in - shifted_target|.
  const int gx = blockIdx.x * TX + threadIdx.x;
  const int gy = blockIdx.y * TY + threadIdx.y;
  const int base = (n * NC) * HWSZ + gy * WW + gx;
  const float i0 = inputs[base];
  const float i1 = inputs[base + HWSZ];
  const float i2 = inputs[base + 2 * HWSZ];

  float best = 3.402823466e+38f;
#pragma unroll
  for (int i = 0; i < 7; ++i) {
#pragma unroll
    for (int j = 0; j < 7; ++j) {
      const int ly = threadIdx.y + i;
      const int lx = threadIdx.x + j;
      float s = fabsf(i0 - tile[0][ly][lx])
              + fabsf(i1 - tile[1][ly][lx])
              + fabsf(i2 - tile[2][ly][lx]);
      best = fminf(best, s);
    }
  }
  best *= (1.0f / 3.0f);   // channel mean

  // Block reduction: WMMA wave32 sum per wave, then 8 wave sums by thread 0.
  const int wid  = tid >> 5;
  const int lane = tid & 31;
  float wsum = wave32_sum_wmma(best);
  if (lane == 0) wpart[wid] = wsum;
  __syncthreads();
  if (tid == 0) {
    float s = 0.0f;
#pragma unroll
    for (int k = 0; k < TX * TY / 32; ++k) s += wpart[k];
    const int bid = blockIdx.x + gridDim.x * (blockIdx.y + gridDim.y * blockIdx.z);
    block_partials[bid] = s;
  }
}

__global__ void __launch_bounds__(32)
final_reduce_kernel(const float* __restrict__ block_partials,
                    int nblocks,
                    float* __restrict__ out) {
  // Single wave32: deterministic strided accumulation, then WMMA wave sum.
  const int lane = threadIdx.x;
  float acc = 0.0f;
  for (int i = lane; i < nblocks; i += 32) acc += block_partials[i];
  float total = wave32_sum_wmma(acc);
  if (lane == 0)
    out[0] = total * (1.0f / (float)(NB * HH * WW));   // mean over N*H*W pixels
}

extern "C" void kernel_launch(void* const* d_in, const int* in_sizes, int n_in,
                              void* d_out, int out_size, void* d_ws, size_t ws_size,
                              hipStream_t stream) {
  (void)in_sizes; (void)n_in; (void)out_size; (void)ws_size;
  const float* inputs  = (const float*)d_in[0];
  const float* targets = (const float*)d_in[1];
  float* out      = (float*)d_out;
  float* partials = (float*)d_ws;   // 16384 floats = 64 KB of scratch

  dim3 grid(WW / TX, HH / TY, NB);  // (16, 64, 16) = 16384 blocks
  dim3 block(TX, TY, 1);            // 256 threads = 8 waves (wave32)
  local_l1_min_kernel<<<grid, block, 0, stream>>>(inputs, targets, partials);

  const int nblocks = (WW / TX) * (HH / TY) * NB;
  final_reduce_kernel<<<1, 32, 0, stream>>>(partials, nblocks, out);
}